// EchoStateNetwork_40991167873519
// MI455X (gfx1250) — compile-verified
//
#include <hip/hip_runtime.h>
#include <hip/hip_bf16.h>

typedef __attribute__((ext_vector_type(2))) float v2f;
typedef __attribute__((ext_vector_type(8))) float v8f;

#define IN_SZ   128
#define HID     1024
#define OUT_SZ  128
#define BATCH   64
#define TSTEPS  1024

// ---------------------------------------------------------------------------
// Zero-init h0 (workspace is poisoned 0xAA; must be re-zeroed every call).
// ---------------------------------------------------------------------------
__global__ __launch_bounds__(256) void esn_zero(float* __restrict__ p, int n) {
    for (int i = blockIdx.x * blockDim.x + threadIdx.x; i < n;
         i += gridDim.x * blockDim.x)
        p[i] = 0.0f;
}

// ---------------------------------------------------------------------------
// FP32 WMMA 16x16 tile over K range [k0,k1) (multiple of 8), with TWO
// independent accumulator chains to halve the dependent-WMMA stall depth.
// A: row-major (lda), tile row base applied. B: row-major (ldb), col base
// applied. ISA layout: A lanes 0-15 hold K+{0,1}, lanes 16-31 hold K+{2,3};
// B mirrored with lane -> N.
// ---------------------------------------------------------------------------
__device__ __forceinline__ v8f wmma_kacc(const float* __restrict__ A, int lda,
                                         const float* __restrict__ B, int ldb,
                                         int k0, int k1, int lane) {
    const int r    = lane & 15;
    const int koff = (lane >> 4) << 1;  // 0 (lanes 0-15) or 2 (lanes 16-31)
    v8f c0 = {}, c1 = {};
    #pragma unroll 2
    for (int k = k0; k < k1; k += 8) {
        v2f a0 = *(const v2f*)(A + r * lda + k + koff);
        v2f b0;
        b0.x = B[(k + koff) * ldb + r];
        b0.y = B[(k + koff + 1) * ldb + r];
        c0 = __builtin_amdgcn_wmma_f32_16x16x4_f32(false, a0, false, b0,
                                                   (short)0, c0, false, false);
        v2f a1 = *(const v2f*)(A + r * lda + (k + 4) + koff);
        v2f b1;
        b1.x = B[(k + 4 + koff) * ldb + r];
        b1.y = B[(k + 5 + koff) * ldb + r];
        c1 = __builtin_amdgcn_wmma_f32_16x16x4_f32(false, a1, false, b1,
                                                   (short)0, c1, false, false);
    }
    #pragma unroll
    for (int i = 0; i < 8; ++i) c0[i] += c1[i];
    return c0;
}

// ---------------------------------------------------------------------------
// Software-pipelined per-step kernel, 289 blocks x 128 threads:
//   blocks [0,256):   recurrence  h_{t+1} = tanh(Wxh[x[:,t]] + b + h_t @ Whh)
//   blocks [256,288): logit tiles for output index s=t-1 (z_s = h_t):
//                     ring[s&3] = h_t @ Wout + bout   (32 tiles, K/4 per wave)
//   block  288:       softmax of ring slot (t-3)&3 -> y[:, t-3, :]
// Ring slot s&3: written at launch s+1, read at launch s+3, reused s+5: safe.
// All WMMA paths have matched depth-64 dependent chains (2x32 after split).
// ---------------------------------------------------------------------------
__global__ __launch_bounds__(128) void esn_fused(
        const float* __restrict__ hp,     // h_t   (stable this launch)
        float*       __restrict__ hn,     // h_{t+1}
        const float* __restrict__ Wrnn,   // 1152 x 1024 (Wxh rows 0..127)
        const float* __restrict__ brnn,   // 1024
        const int*   __restrict__ x,      // 64 x 1024
        const float* __restrict__ Wout,   // 1024 x 128
        const float* __restrict__ bout,   // 128
        float*       __restrict__ ring,   // 4 x 64 x 128 logits ring
        float*       __restrict__ y,      // 64 x 1024 x 128
        int t) {
    __shared__ float part[4][8][32];
    __shared__ float red[2][2][64];       // [max/sum][half][row]

    const int bid  = blockIdx.x;
    const int lane = threadIdx.x & 31;
    const int w    = threadIdx.x >> 5;

    if (bid < 256) {
        // ----- recurrence step t -----
        if (t >= TSTEPS) return;
        const int mt = bid >> 6;          // 0..3  M tile
        const int nt = bid & 63;          // 0..63 N tile
        const float* Whh = Wrnn + IN_SZ * HID;
        const float* A   = hp + mt * 16 * HID;
        const float* B   = Whh + nt * 16;

        v8f c = wmma_kacc(A, HID, B, HID, w * 256, (w + 1) * 256, lane);
        #pragma unroll
        for (int i = 0; i < 8; ++i) part[w][i][lane] = c[i];
        __syncthreads();

        // epilogue spread over all 4 waves: wave w handles i = 2w, 2w+1
        const int col  = nt * 16 + (lane & 15);
        const float bb = brnn[col];
        #pragma unroll
        for (int ii = 0; ii < 2; ++ii) {
            const int i   = 2 * w + ii;
            const int m   = i + ((lane >> 4) << 3);
            const int row = mt * 16 + m;             // batch index
            float s = part[0][i][lane] + part[1][i][lane] +
                      part[2][i][lane] + part[3][i][lane];
            const int idx = x[row * TSTEPS + t];
            s += Wrnn[idx * HID + col] + bb;         // Wxh gather + b_rnn
            hn[row * HID + col] = tanhf(s);
        }
    } else if (bid < 288) {
        // ----- logits for output index s = t-1 : ring[s&3] = h_t@Wout+bout --
        const int s = t - 1;
        if (s < 0 || s >= TSTEPS) return;
        const int id = bid - 256;         // 0..31
        const int mt = id >> 3;           // 0..3
        const int nt = id & 7;            // 0..7
        const float* A = hp + mt * 16 * HID;
        const float* B = Wout + nt * 16;

        v8f c = wmma_kacc(A, HID, B, OUT_SZ, w * 256, (w + 1) * 256, lane);
        #pragma unroll
        for (int i = 0; i < 8; ++i) part[w][i][lane] = c[i];
        __syncthreads();

        float* slot = ring + (size_t)(s & 3) * BATCH * OUT_SZ;
        const int col = nt * 16 + (lane & 15);
        const float bb = bout[col];
        #pragma unroll
        for (int ii = 0; ii < 2; ++ii) {
            const int i   = 2 * w + ii;
            const int m   = i + ((lane >> 4) << 3);
            const int row = mt * 16 + m;
            float v = part[0][i][lane] + part[1][i][lane] +
                      part[2][i][lane] + part[3][i][lane];
            slot[row * OUT_SZ + col] = v + bb;
        }
    } else {
        // ----- softmax of ring slot for output index s = t-3 -----
        const int s = t - 3;
        if (s < 0 || s >= TSTEPS) return;
        const float* slot = ring + (size_t)(s & 3) * BATCH * OUT_SZ;
        const int row  = threadIdx.x & 63;          // batch row
        const int half = threadIdx.x >> 6;          // 0/1: cols [0,64)/[64,128)
        const float* v = slot + row * OUT_SZ + half * 64;

        float mx = -3.402823466e+38f;
        for (int j = 0; j < 64; ++j) mx = fmaxf(mx, v[j]);
        red[0][half][row] = mx;
        __syncthreads();
        mx = fmaxf(red[0][0][row], red[0][1][row]);

        float sum = 0.0f;
        for (int j = 0; j < 64; ++j) sum += expf(v[j] - mx);
        red[1][half][row] = sum;
        __syncthreads();
        const float inv = 1.0f / (red[1][0][row] + red[1][1][row]);

        float* yo = y + (size_t)row * TSTEPS * OUT_SZ + (size_t)s * OUT_SZ +
                    half * 64;
        for (int j = 0; j < 64; ++j) yo[j] = expf(v[j] - mx) * inv;
    }
}

// ---------------------------------------------------------------------------
// Driver. d_in order: x (int32 64x1024), W_rnn (1152x1024), b_rnn (1024),
// W_out (1024x128), b_out (128).
// d_out: y (64*1024*128 floats) then hT (64*1024 floats).
// ws: h ping (256KB) | h pong (256KB) | logits ring 4x64x128 (128KB).
// ---------------------------------------------------------------------------
extern "C" void kernel_launch(void* const* d_in, const int* in_sizes, int n_in,
                              void* d_out, int out_size, void* d_ws, size_t ws_size,
                              hipStream_t stream) {
    (void)in_sizes; (void)n_in; (void)out_size; (void)ws_size;

    const int*   x    = (const int*)d_in[0];
    const float* Wrnn = (const float*)d_in[1];
    const float* brnn = (const float*)d_in[2];
    const float* Wout = (const float*)d_in[3];
    const float* bout = (const float*)d_in[4];

    float* y  = (float*)d_out;
    float* hT = (float*)d_out + (size_t)BATCH * TSTEPS * OUT_SZ;

    float* h0   = (float*)d_ws;                    // h_t for even t
    float* h1   = h0 + BATCH * HID;                // h_t for odd t
    float* ring = h1 + BATCH * HID;                // 4-slot logits ring

    esn_zero<<<64, 256, 0, stream>>>(h0, BATCH * HID);

    // t in [0, TSTEPS+2]: recurrence drains at t=TSTEPS-1, logits at
    // t=TSTEPS, softmax at t=TSTEPS+2.
    for (int t = 0; t <= TSTEPS + 2; ++t) {
        const float* hp = (t & 1) ? h1 : h0;       // h_t lives in buf[t&1]
        float*       hn = (t & 1) ? h0 : h1;
        esn_fused<<<289, 128, 0, stream>>>(hp, hn, Wrnn, brnn, x,
                                           Wout, bout, ring, y, t);
    }

    // h_{1024} lives in buf[1024 & 1] = h0.
    (void)hipMemcpyAsync(hT, h0, (size_t)BATCH * HID * sizeof(float),
                         hipMemcpyDeviceToDevice, stream);
}